// MoPeAttention_38517266710608
// MI455X (gfx1250) — compile-verified
//
#include <hip/hip_runtime.h>
#include <math.h>

// ---------------------------------------------------------------------------
// MoPeAttention for MI455X (gfx1250, wave32).
//   K0 sig_kernel      : sigmoid of lerp weights (5 x F table)
//   K1 ln_kernel       : LayerNorm of concat(last_x, x)   (B*(T+1) rows)
//   K2 proj_wmma_kernel: 5 fused lerp-linear projections via V_WMMA_F32_16X16X4_F32
//   K3 scan_kernel     : elementwise linear recurrence over T, double-buffered
//                        LDS staging via GLOBAL_LOAD_ASYNC_TO_LDS_B32 (ASYNCcnt)
//   K4 solve_kernel    : per (b,t): Gauss-Jordan solve A z = q, out_x = B z
//   K5 out_wmma_kernel : y = x + out_x @ out_w^T + out_b via WMMA f32
// ---------------------------------------------------------------------------

typedef float v2f __attribute__((ext_vector_type(2)));
typedef float v8f __attribute__((ext_vector_type(8)));

#define BB 4
#define TT 1024
#define FF 512
#define KK 64
#define CC 448                    // q(64) k(64) v(64) r_k(64) r_v(64) w_k(64) w_v(64)
#define NROWS (BB*TT)             // 4096
#define LN_EPS 1e-5f

__device__ __forceinline__ float sigf(float x) { return 1.0f / (1.0f + __expf(-x)); }

// Select the (K = 2*kh, 2*kh+1) pair of a float4 K-slab per ISA 16x4 f32 A/B layout.
__device__ __forceinline__ v2f pick2(float4 v, int kh) {
    v2f r;
    if (kh) { r.x = v.z; r.y = v.w; } else { r.x = v.x; r.y = v.y; }
    return r;
}

// Async copy of one DWORD global -> LDS, tracked by ASYNCcnt (no VGPR data path).
__device__ __forceinline__ void async_g2l_b32(unsigned lds_off, const float* gaddr) {
    asm volatile("global_load_async_to_lds_b32 %0, %1, off"
                 :: "v"(lds_off), "v"(gaddr) : "memory");
}
__device__ __forceinline__ void wait_asynccnt0() {
    asm volatile("s_wait_asynccnt 0x0" ::: "memory");
}

// ---------------- K0: sigmoid tables for the 5 lerp weight vectors ----------
__global__ void sig_kernel(const float* __restrict__ qkv_lerp,
                           const float* __restrict__ rw_lerp,
                           float* __restrict__ sig) {
    int idx = blockIdx.x * blockDim.x + threadIdx.x;
    if (idx >= 5 * FF) return;
    int i = idx / FF, f = idx % FF;
    float v = (i < 3) ? qkv_lerp[i * FF + f] : rw_lerp[(i - 3) * FF + f];
    sig[idx] = sigf(v);
}

// ---------------- K1: LayerNorm over F on concat(last_x, x) -----------------
__global__ void ln_kernel(const float* __restrict__ x,
                          const float* __restrict__ last_x,
                          const float* __restrict__ gamma,
                          const float* __restrict__ beta,
                          float* __restrict__ norm) {
    int row = blockIdx.x;                       // 0 .. B*(T+1)-1
    int b = row / (TT + 1), tt = row % (TT + 1);
    const float* src = (tt == 0) ? (last_x + (size_t)b * FF)
                                 : (x + ((size_t)b * TT + (tt - 1)) * FF);
    int tid = threadIdx.x;                      // 256 threads, 2 elems each
    float a0 = src[tid], a1 = src[tid + 256];
    __shared__ float red[256];

    red[tid] = a0 + a1;
    __syncthreads();
    for (int s = 128; s > 0; s >>= 1) { if (tid < s) red[tid] += red[tid + s]; __syncthreads(); }
    float mean = red[0] * (1.0f / FF);
    __syncthreads();

    red[tid] = a0 * a0 + a1 * a1;
    __syncthreads();
    for (int s = 128; s > 0; s >>= 1) { if (tid < s) red[tid] += red[tid + s]; __syncthreads(); }
    float var = red[0] * (1.0f / FF) - mean * mean;
    float rstd = rsqrtf(var + LN_EPS);

    float* dst = norm + (size_t)row * FF;
    dst[tid]       = (a0 - mean) * rstd * gamma[tid]       + beta[tid];
    dst[tid + 256] = (a1 - mean) * rstd * gamma[tid + 256] + beta[tid + 256];
}

// ---------------- K2: fused lerp + projection GEMM (WMMA f32 16x16x4) -------
// proj[row][c], c: [0,64)=q [64,128)=k [128,192)=v [192,320)=r [320,448)=w
__global__ void proj_wmma_kernel(const float* __restrict__ norm,
                                 const float* __restrict__ qkv_w,
                                 const float* __restrict__ rw_w,
                                 const float* __restrict__ sig,
                                 float* __restrict__ proj) {
    int wave = threadIdx.x >> 5;                // 4 waves / block
    int lane = threadIdx.x & 31;
    int ntile = blockIdx.y * 4 + wave;          // 0..27  (28 N-tiles of 16)
    int n0 = ntile * 16;
    int M0 = blockIdx.x * 16;                   // 256 M-tiles of 16 rows
    int ln15 = lane & 15;                       // row m (A) / col n (B)
    int kh   = lane >> 4;                       // K-half select

    int row = M0 + ln15;                        // flat (b,t)
    int b = row >> 10, t = row & 1023;
    const float* nl = norm + ((size_t)b * (TT + 1) + t) * FF;  // norm_last
    const float* nc = nl + FF;                                  // norm_curr

    const float* wrow = (n0 < 192) ? (qkv_w + (size_t)(n0 + ln15) * FF)
                                   : (rw_w  + (size_t)(n0 - 192 + ln15) * FF);
    int pi = (n0 < 64) ? 0 : (n0 < 128) ? 1 : (n0 < 192) ? 2 : (n0 < 320) ? 3 : 4;
    const float* sg = sig + (size_t)pi * FF;

    v8f acc = {};
    for (int kk = 0; kk < FF; kk += 4) {
        float4 l4 = *(const float4*)(nl + kk);
        float4 c4 = *(const float4*)(nc + kk);
        float4 s4 = *(const float4*)(sg + kk);
        float4 lin;
        lin.x = l4.x + (c4.x - l4.x) * s4.x;
        lin.y = l4.y + (c4.y - l4.y) * s4.y;
        lin.z = l4.z + (c4.z - l4.z) * s4.z;
        lin.w = l4.w + (c4.w - l4.w) * s4.w;
        float4 w4 = *(const float4*)(wrow + kk);
        v2f a = pick2(lin, kh);                 // A[m][2kh..2kh+1]
        v2f bm = pick2(w4, kh);                 // B[2kh..2kh+1][n]
        acc = __builtin_amdgcn_wmma_f32_16x16x4_f32(
            false, a, false, bm, (short)0, acc, false, false);
    }

    int act = (n0 >= 320) ? 2 : (n0 >= 192 ? 1 : 0);
    for (int i = 0; i < 8; ++i) {
        int m = i + 8 * kh;                     // C/D layout: VGPR i -> row i + 8*kh
        float v = acc[i];
        if (act == 1)      v = 0.99f + 0.01f * sigf(v);                        // r
        else if (act == 2) v = (v >= 0.0f) ? (v + 1.0f) : (1.0f / (1.0f - v)); // w = _s
        proj[(size_t)(M0 + m) * CC + n0 + ln15] = v;
    }
}

// ---------------- K3: elementwise recurrence over T -------------------------
// A[k,l] <- A*R + (k_t[k]*k_t[l])*W ;  B[k,l] <- B*R + (v_t[k]*k_t[l])*W
// Double-buffered LDS staging with GLOBAL_LOAD_ASYNC_TO_LDS_B32 so the t+1
// fetch overlaps the t compute + history stores.  Block = 4 k-rows x 64 l.
__global__ void scan_kernel(const float* __restrict__ proj,
                            const float* __restrict__ last_A,
                            const float* __restrict__ last_B,
                            float* __restrict__ Ahist,
                            float* __restrict__ Bhist,
                            float* __restrict__ outA,
                            float* __restrict__ outB) {
    int b = blockIdx.y;
    int tid = threadIdx.x;                      // 256
    int k = blockIdx.x * 4 + (tid >> 6);
    int l = tid & 63;
    float A  = last_A[k * KK + l];
    float Bm = last_B[k * KK + l];
    __shared__ float s[2][384];                 // k | v | r_k | r_v | w_k | w_v

    // prime buffer 0 with t = 0
    {
        const float* pr = proj + ((size_t)(b * TT)) * CC + KK;
        async_g2l_b32((unsigned)(size_t)&s[0][tid], pr + tid);
        if (tid < 128)
            async_g2l_b32((unsigned)(size_t)&s[0][256 + tid], pr + 256 + tid);
    }

    for (int t = 0; t < TT; ++t) {
        int cur = t & 1, nxt = cur ^ 1;
        wait_asynccnt0();                       // my wave's async copies landed
        __syncthreads();                        // everyone's copies visible
        float kl  = s[cur][l];
        float kkv = s[cur][k];
        float vk  = s[cur][64 + k];
        float R   = s[cur][192 + k] * s[cur][128 + l];   // r_v[k]*r_k[l]
        float W   = s[cur][320 + k] * s[cur][256 + l];   // w_v[k]*w_k[l]
        __syncthreads();                        // all reads done before nxt refill
        if (t + 1 < TT) {                       // overlap t+1 fetch with compute
            const float* pr = proj + ((size_t)(b * TT + t + 1)) * CC + KK;
            async_g2l_b32((unsigned)(size_t)&s[nxt][tid], pr + tid);
            if (tid < 128)
                async_g2l_b32((unsigned)(size_t)&s[nxt][256 + tid], pr + 256 + tid);
        }
        A  = A  * R + kkv * kl * W;             // dA = k[k]*k[l]
        Bm = Bm * R + vk  * kl * W;             // dB = v[k]*k[l]
        size_t o = ((size_t)(b * TT + t)) * (KK * KK) + k * KK + l;
        Ahist[o] = A;
        Bhist[o] = Bm;
    }
    outA[(size_t)b * KK * KK + k * KK + l] = A;
    outB[(size_t)b * KK * KK + k * KK + l] = Bm;
}

// ---------------- K4: per-(b,t) single-RHS solve + B@z ----------------------
// out_x = B_t * (A_t^{-1} q_t): Gauss-Jordan (no pivot; A is SPD-dominant).
__global__ void solve_kernel(const float* __restrict__ Ahist,
                             const float* __restrict__ Bhist,
                             const float* __restrict__ proj,
                             float* __restrict__ outx) {
    int row = blockIdx.x;                       // 0..4095
    int tid = threadIdx.x;                      // 256
    __shared__ float M[KK * 65];                // augmented [A | q]

    const float* Ar = Ahist + (size_t)row * (KK * KK);
    for (int e = tid; e < KK * KK; e += 256) {
        int r = e >> 6, c = e & 63;
        M[r * 65 + c] = Ar[e];
    }
    if (tid < KK) M[tid * 65 + 64] = proj[(size_t)row * CC + tid];   // q
    __syncthreads();

    for (int p = 0; p < KK; ++p) {
        float pinv = 1.0f / M[p * 65 + p];
        int c = p + tid;
        if (c <= 64) M[p * 65 + c] *= pinv;     // normalize pivot row
        __syncthreads();
        for (int e = tid; e < KK * 65; e += 256) {
            int r = e / 65, c2 = e % 65;
            if (r != p && c2 > p)
                M[e] -= M[r * 65 + p] * M[p * 65 + c2];
        }
        __syncthreads();
    }

    if (tid < KK) {                             // out_x[k] = sum_l B[k,l] * z[l]
        const float* Br = Bhist + (size_t)row * (KK * KK) + (size_t)tid * KK;
        float acc = 0.0f;
        for (int l = 0; l < KK; ++l) acc += Br[l] * M[l * 65 + 64];
        outx[(size_t)row * KK + tid] = acc;
    }
}

// ---------------- K5: output projection + residual (WMMA f32) ---------------
__global__ void out_wmma_kernel(const float* __restrict__ outx,
                                const float* __restrict__ out_w,
                                const float* __restrict__ out_b,
                                const float* __restrict__ x,
                                float* __restrict__ y) {
    int wave = threadIdx.x >> 5;
    int lane = threadIdx.x & 31;
    int ntile = blockIdx.y * 4 + wave;          // 32 N-tiles of 16 (F=512)
    int n0 = ntile * 16;
    int M0 = blockIdx.x * 16;
    int ln15 = lane & 15;
    int kh = lane >> 4;

    const float* arow = outx  + (size_t)(M0 + ln15) * KK;
    const float* wrow = out_w + (size_t)(n0 + ln15) * KK;   // out_w is (F,K)
    v8f acc = {};
    for (int kk = 0; kk < KK; kk += 4) {
        v2f a  = pick2(*(const float4*)(arow + kk), kh);
        v2f bm = pick2(*(const float4*)(wrow + kk), kh);
        acc = __builtin_amdgcn_wmma_f32_16x16x4_f32(
            false, a, false, bm, (short)0, acc, false, false);
    }
    int f = n0 + ln15;
    float bias = out_b[f];
    for (int i = 0; i < 8; ++i) {
        int m = i + 8 * kh;
        size_t o = (size_t)(M0 + m) * FF + f;
        y[o] = x[o] + acc[i] + bias;
    }
}

// ---------------------------------------------------------------------------
extern "C" void kernel_launch(void* const* d_in, const int* in_sizes, int n_in,
                              void* d_out, int out_size, void* d_ws, size_t ws_size,
                              hipStream_t stream) {
    const float* x        = (const float*)d_in[0];
    const float* last_x   = (const float*)d_in[1];
    const float* last_A   = (const float*)d_in[2];
    const float* last_B   = (const float*)d_in[3];
    const float* ln_gamma = (const float*)d_in[4];
    const float* ln_beta  = (const float*)d_in[5];
    const float* qkv_lerp = (const float*)d_in[6];
    const float* qkv_w    = (const float*)d_in[7];
    const float* rw_lerp  = (const float*)d_in[8];
    const float* rw_w     = (const float*)d_in[9];
    const float* out_w    = (const float*)d_in[10];
    const float* out_b    = (const float*)d_in[11];

    float* ws = (float*)d_ws;
    size_t off = 0;
    float* sig  = ws + off; off += (size_t)5 * FF;                 // 2,560
    float* norm = ws + off; off += (size_t)BB * (TT + 1) * FF;     // 2,099,200
    float* proj = ws + off; off += (size_t)NROWS * CC;             // 1,835,008
    float* Ah   = ws + off; off += (size_t)NROWS * KK * KK;        // 16,777,216
    float* Bh   = ws + off; off += (size_t)NROWS * KK * KK;        // 16,777,216
    float* ox   = ws + off; off += (size_t)NROWS * KK;             // 262,144
    (void)ws_size; (void)in_sizes; (void)n_in; (void)out_size;

    float* y    = (float*)d_out;                                   // (B,T,F)
    float* outA = y + (size_t)NROWS * FF;                          // (B,K,K)
    float* outB = outA + (size_t)BB * KK * KK;                     // (B,K,K)

    sig_kernel<<<10, 256, 0, stream>>>(qkv_lerp, rw_lerp, sig);
    ln_kernel<<<BB * (TT + 1), 256, 0, stream>>>(x, last_x, ln_gamma, ln_beta, norm);
    proj_wmma_kernel<<<dim3(NROWS / 16, 7), 128, 0, stream>>>(norm, qkv_w, rw_w, sig, proj);
    scan_kernel<<<dim3(16, BB), 256, 0, stream>>>(proj, last_A, last_B, Ah, Bh, outA, outB);
    solve_kernel<<<NROWS, 256, 0, stream>>>(Ah, Bh, proj, ox);
    out_wmma_kernel<<<dim3(NROWS / 16, 8), 128, 0, stream>>>(ox, out_w, out_b, x, y);
}